// PerformerFeatureMap_27496380629322
// MI455X (gfx1250) — compile-verified
//
#include <hip/hip_runtime.h>

// Performer FAVOR+ feature map for MI455X (gfx1250, wave32, WMMA).
//   out[t, j] = 2^-4 * ( exp( 64^-0.25 * dot(x[t,:], proj[j,:]) - 0.5*||x[t,:]||^2 ) + 1e-4 )
// Single flat GEMM [131072,64] x [64,256] with fused epilogue, fp32 WMMA 16x16x4.

typedef __attribute__((ext_vector_type(2))) float v2f;
typedef __attribute__((ext_vector_type(8))) float v8f;

static constexpr int HD        = 64;    // head dim (K)
static constexpr int NF        = 256;   // nb_features (N)
static constexpr int ROWS_WG   = 128;   // output rows per workgroup
static constexpr int THREADS   = 256;   // 8 waves
static constexpr int LDS_STR   = 68;    // 64 + 4 pad: conflict-free frag reads

__global__ __launch_bounds__(THREADS)
void performer_featuremap_kernel(const float* __restrict__ x,
                                 const float* __restrict__ proj,
                                 float* __restrict__ out)
{
    __shared__ float lds_x[ROWS_WG * LDS_STR];   //  34 KB
    __shared__ float lds_p[NF      * LDS_STR];   //  68 KB

    const int tid  = threadIdx.x;
    const int wave = tid >> 5;
    const int lane = tid & 31;
    const int n    = lane & 15;   // row-in-A-tile / col-in-B/D-tile
    const int half = lane >> 4;   // 0 or 1

    const long rowbase = (long)blockIdx.x * ROWS_WG;

    // ---- cooperative load: x tile (128 x 64 fp32), coalesced b128 ----
    {
        const float4* x4 = (const float4*)(x + rowbase * HD);
        #pragma unroll
        for (int i = tid; i < ROWS_WG * (HD / 4); i += THREADS) {
            const int r = i >> 4, c4 = i & 15;
            float4 v = x4[i];
            *(float4*)&lds_x[r * LDS_STR + c4 * 4] = v;
        }
    }
    // ---- cooperative load: projection (256 x 64 fp32) ----
    {
        const float4* p4 = (const float4*)proj;
        #pragma unroll
        for (int i = tid; i < NF * (HD / 4); i += THREADS) {
            const int r = i >> 4, c4 = i & 15;
            float4 v = p4[i];
            *(float4*)&lds_p[r * LDS_STR + c4 * 4] = v;
        }
    }
    __syncthreads();

    // ---- diag = 0.5*||x_row||^2 for this wave's 16 rows ----
    float ss = 0.0f;
    {
        const float* rp = &lds_x[(wave * 16 + n) * LDS_STR + half * 32];
        #pragma unroll
        for (int c = 0; c < 8; ++c) {
            float4 v = *(const float4*)&rp[c * 4];
            ss += v.x * v.x + v.y * v.y + v.z * v.z + v.w * v.w;
        }
    }
    ss += __shfl_xor(ss, 16, 32);          // combine the two half-row sums
    const float diagval = 0.5f * ss;       // lane L holds diag of row L%16
    float d8[8];                           // diag in D-matrix layout: VGPR i -> row i+8*half
    #pragma unroll
    for (int i = 0; i < 8; ++i)
        d8[i] = __shfl(diagval, i + 8 * half, 32);

    // ---- A fragments: 16 k-steps of the wave's 16x64 slab ----
    // V_WMMA_F32_16X16X4_F32 A layout: lane L = row L%16;
    //   VGPR0 = K = 4k + 2*(L/16), VGPR1 = K+1  -> one float2 per k-step.
    float a0[16], a1[16];
    {
        const float* ap = &lds_x[(wave * 16 + n) * LDS_STR + 2 * half];
        #pragma unroll
        for (int k = 0; k < 16; ++k) {
            float2 v = *(const float2*)&ap[4 * k];
            a0[k] = v.x; a1[k] = v.y;
        }
    }

    const float normalizer = 0.35355339059327379f;  // 64^(-1/4)
    const float ratio      = 0.0625f;               // 256^(-1/2)

    float* outw = out + (rowbase + wave * 16) * NF;

    #pragma unroll 1
    for (int jt = 0; jt < 16; ++jt) {               // 16 column tiles of 16
        v8f acc = {};
        const float* bp = &lds_p[(jt * 16 + n) * LDS_STR + 2 * half];
        #pragma unroll
        for (int k = 0; k < 16; ++k) {              // K = 64 in steps of 4
            float2 bv = *(const float2*)&bp[4 * k];
            v2f A; A.x = a0[k]; A.y = a1[k];
            v2f B; B.x = bv.x;  B.y = bv.y;
            acc = __builtin_amdgcn_wmma_f32_16x16x4_f32(
                      /*neg_a=*/false, A, /*neg_b=*/false, B,
                      /*c_mod=*/(short)0, acc, /*reuse_a=*/false, /*reuse_b=*/false);
        }
        // Epilogue: D layout VGPR i = row (i + 8*half), col = jt*16 + n
        #pragma unroll
        for (int i = 0; i < 8; ++i) {
            float dd = normalizer * acc[i] - d8[i];
            float v  = ratio * (__expf(dd) + 1e-4f);
            outw[(i + 8 * half) * NF + jt * 16 + n] = v;
        }
    }
}

extern "C" void kernel_launch(void* const* d_in, const int* in_sizes, int n_in,
                              void* d_out, int out_size, void* d_ws, size_t ws_size,
                              hipStream_t stream) {
    const float* x    = (const float*)d_in[0];
    const float* proj = (const float*)d_in[1];
    float* out        = (float*)d_out;

    const long tokens = (long)in_sizes[0] / HD;       // 2*16*4096 = 131072
    const int  blocks = (int)(tokens / ROWS_WG);      // 1024

    hipLaunchKernelGGL(performer_featuremap_kernel,
                       dim3(blocks), dim3(THREADS), 0, stream,
                       x, proj, out);
}